// ZBLRepulsion_1211180777472
// MI455X (gfx1250) — compile-verified
//
#include <hip/hip_runtime.h>
#include <math.h>

// ---------------------------------------------------------------------------
// ZBL repulsion: edge-wise gather + screened-Coulomb energy + global sum.
// MI455X plan:
//  - idx rows streamed as int4 (global_load_b128): 51.2 MB once = HBM floor
//    (~2.2us @ 23.3 TB/s).
//  - R (1.2 MB) / Z (0.4 MB) gathered from L2/WGP$ (192 MB L2 -> resident).
//  - Z^a_exp via 64-entry LDS table (Z is int in [1,50)) -> kills 4
//    transcendentals per edge; v_rcp_f32 instead of IEEE divide.
//  - Wave reduction via V_WMMA_F32_16X16X4_F32 (A = lane partials, B = ones).
// ---------------------------------------------------------------------------

typedef __attribute__((ext_vector_type(2))) float v2f;
typedef __attribute__((ext_vector_type(8))) float v8f;

#define ZBL_RMAX 6.0f
#define ZBL_PI_OVER_RMAX 0.52359877559829887f  // pi / 6.0
#define ZTAB_SIZE 64

// Wave+block reduction. Each lane contributes (acc0, acc1) as one 2-element
// slice of the 16x4 A matrix; B = all-ones 4x16. D[m][n] = rowsum_m(A).
// Per the CDNA5 16x16 f32 C/D layout (8 VGPRs x 32 lanes): lanes 0-15 hold
// rowsums m=0..7 across their 8 VGPRs, lanes 16-31 hold m=8..15. Summing a
// lane's 8 VGPRs and one xor-16 shuffle yields sum(A) = the wave partial.
// Returns the block sum on thread 0 (undefined elsewhere).
__device__ __forceinline__ float block_reduce_wmma(float acc0, float acc1,
                                                   float* warp_sums) {
    v2f a; a.x = acc0; a.y = acc1;
    v2f b; b.x = 1.0f; b.y = 1.0f;
    v8f c = {};
    // (neg_a, A, neg_b, B, c_mod, C, reuse_a, reuse_b)
    v8f d = __builtin_amdgcn_wmma_f32_16x16x4_f32(
        false, a, false, b, (short)0, c, false, false);
    float s = d[0] + d[1] + d[2] + d[3] + d[4] + d[5] + d[6] + d[7];
    s += __shfl_xor(s, 16, 32);            // combine half-waves -> sum(A)

    const int lane = threadIdx.x & 31;
    const int wid  = threadIdx.x >> 5;
    if (lane == 0) warp_sums[wid] = s;
    __syncthreads();

    float blk = 0.0f;
    if (threadIdx.x == 0) {
        const int nw = blockDim.x >> 5;    // 8 wave32s per 256-thread block
        for (int w = 0; w < nw; ++w) blk += warp_sums[w];
    }
    return blk;
}

__device__ __forceinline__ float edge_energy(
    int i, int j,
    const float* __restrict__ R, const int* __restrict__ Z,
    const float* __restrict__ ztab,        // LDS: z^a_exp, z in [0,64)
    unsigned n_atoms, float inv_anum,
    float c0, float c1, float c2, float c3,
    float x0, float x1, float x2, float x3) {
    // mask_by_neighbor: padded edges carry idx >= n_atoms
    if (((unsigned)i >= n_atoms) || ((unsigned)j >= n_atoms)) return 0.0f;

    const float xi = R[3 * i + 0], yi = R[3 * i + 1], zi = R[3 * i + 2];
    const float xj = R[3 * j + 0], yj = R[3 * j + 1], zj = R[3 * j + 2];
    const float dx = xj - xi, dy = yj - yi, dz = zj - zi;

    float dr = sqrtf(dx * dx + dy * dy + dz * dz);
    dr = fminf(fmaxf(dr, 0.02f), ZBL_RMAX);

    const float cosc = 0.5f * (__cosf(ZBL_PI_OVER_RMAX * dr) + 1.0f);

    const int zi_n = Z[i];
    const int zj_n = Z[j];
    const unsigned zi_c =
        ((unsigned)zi_n < (unsigned)ZTAB_SIZE) ? (unsigned)zi_n : ZTAB_SIZE - 1u;
    const unsigned zj_c =
        ((unsigned)zj_n < (unsigned)ZTAB_SIZE) ? (unsigned)zj_n : ZTAB_SIZE - 1u;
    const float adiv = ztab[zi_c] + ztab[zj_c];
    const float dist = dr * adiv * inv_anum;

    const float f = c0 * __expf(-x0 * dist) + c1 * __expf(-x1 * dist) +
                    c2 * __expf(-x2 * dist) + c3 * __expf(-x3 * dist);

    // dr in [0.02, 6] -> v_rcp_f32 is safe; avoids IEEE div fixup chain.
    return 0.5f * (float)zi_n * (float)zj_n * __builtin_amdgcn_rcpf(dr) * f *
           cosc;
}

__global__ void __launch_bounds__(256)
zbl_edge_kernel(const float* __restrict__ R, const int* __restrict__ Z,
                const int* __restrict__ idx,
                const float* __restrict__ a_exp, const float* __restrict__ a_num,
                const float* __restrict__ coef, const float* __restrict__ expo,
                float* __restrict__ partials, int n_atoms, long long n_edges) {
    __shared__ float warp_sums[8];
    __shared__ float ztab[ZTAB_SIZE];

    // Loop-invariant scalars (uniform -> SGPRs).
    const float aexp     = a_exp[0];
    const float inv_anum = 1.0f / a_num[0];
    const float c0 = coef[0], c1 = coef[1], c2 = coef[2], c3 = coef[3];
    const float x0 = expo[0], x1 = expo[1], x2 = expo[2], x3 = expo[3];

    // Build the z^a_exp table once per block (Z is an integer in [1,50)).
    if (threadIdx.x < ZTAB_SIZE) {
        ztab[threadIdx.x] = __powf((float)threadIdx.x, aexp);
    }
    __syncthreads();

    const int* idx_i = idx;
    const int* idx_j = idx + n_edges;

    const unsigned long long tid =
        (unsigned long long)blockIdx.x * blockDim.x + threadIdx.x;
    const unsigned long long stride =
        (unsigned long long)gridDim.x * blockDim.x;
    const unsigned n_atoms_u = (unsigned)n_atoms;

    float acc0 = 0.0f, acc1 = 0.0f;

    if ((n_edges & 3) == 0) {
        // Vector path: 4 edges / iteration via two global_load_b128 streams.
        const int4* __restrict__ idx_i4 = (const int4*)idx_i;
        const int4* __restrict__ idx_j4 = (const int4*)idx_j;
        const unsigned long long n4 = (unsigned long long)n_edges >> 2;
        for (unsigned long long q = tid; q < n4; q += stride) {
            const int4 vi = idx_i4[q];
            const int4 vj = idx_j4[q];
            acc0 += edge_energy(vi.x, vj.x, R, Z, ztab, n_atoms_u, inv_anum,
                                c0, c1, c2, c3, x0, x1, x2, x3);
            acc1 += edge_energy(vi.y, vj.y, R, Z, ztab, n_atoms_u, inv_anum,
                                c0, c1, c2, c3, x0, x1, x2, x3);
            acc0 += edge_energy(vi.z, vj.z, R, Z, ztab, n_atoms_u, inv_anum,
                                c0, c1, c2, c3, x0, x1, x2, x3);
            acc1 += edge_energy(vi.w, vj.w, R, Z, ztab, n_atoms_u, inv_anum,
                                c0, c1, c2, c3, x0, x1, x2, x3);
        }
    } else {
        // Generic scalar path (unused for E = 6.4M but keeps us correct).
        for (unsigned long long e = tid; e < (unsigned long long)n_edges;
             e += 2ull * stride) {
            acc0 += edge_energy(idx_i[e], idx_j[e], R, Z, ztab, n_atoms_u,
                                inv_anum, c0, c1, c2, c3, x0, x1, x2, x3);
            const unsigned long long e2 = e + stride;
            if (e2 < (unsigned long long)n_edges) {
                acc1 += edge_energy(idx_i[e2], idx_j[e2], R, Z, ztab,
                                    n_atoms_u, inv_anum, c0, c1, c2, c3, x0,
                                    x1, x2, x3);
            }
        }
    }

    const float blk = block_reduce_wmma(acc0, acc1, warp_sums);
    if (threadIdx.x == 0) partials[blockIdx.x] = blk;
}

__global__ void __launch_bounds__(256)
zbl_reduce_kernel(const float* __restrict__ partials, int n,
                  float* __restrict__ out) {
    __shared__ float warp_sums[8];
    float acc0 = 0.0f, acc1 = 0.0f;
    for (int i = threadIdx.x; i < n; i += 2 * (int)blockDim.x) {
        acc0 += partials[i];
        const int i2 = i + (int)blockDim.x;
        if (i2 < n) acc1 += partials[i2];
    }
    const float blk = block_reduce_wmma(acc0, acc1, warp_sums);
    if (threadIdx.x == 0) out[0] = blk;
}

extern "C" void kernel_launch(void* const* d_in, const int* in_sizes, int n_in,
                              void* d_out, int out_size, void* d_ws,
                              size_t ws_size, hipStream_t stream) {
    const float* R     = (const float*)d_in[0];  // [N,3] f32
    const int*   Z     = (const int*)d_in[1];    // [N]   i32
    const int*   idx   = (const int*)d_in[2];    // [2,E] i32, row-major
    const float* a_exp = (const float*)d_in[3];  // [1]
    const float* a_num = (const float*)d_in[4];  // [1]
    const float* coef  = (const float*)d_in[5];  // [4,1]
    const float* expo  = (const float*)d_in[6];  // [4,1]

    const int       n_atoms = in_sizes[1];
    const long long n_edges = (long long)in_sizes[2] / 2;

    float* partials = (float*)d_ws;              // GRID floats (8 KB)

    constexpr int BLOCK = 256;                   // 8 wave32s
    constexpr int GRID  = 2048;                  // ~12 edges/thread

    zbl_edge_kernel<<<GRID, BLOCK, 0, stream>>>(
        R, Z, idx, a_exp, a_num, coef, expo, partials, n_atoms, n_edges);
    zbl_reduce_kernel<<<1, BLOCK, 0, stream>>>(partials, GRID,
                                               (float*)d_out);
}